// My_parse_cnn_layer_11922829214242
// MI455X (gfx1250) — compile-verified
//
#include <hip/hip_runtime.h>

// CDNA5 / gfx1250: wave32, native fp32 WMMA 16x16x4.
typedef __attribute__((ext_vector_type(2))) float v2f;
typedef __attribute__((ext_vector_type(8))) float v8f;

#define Bn 100          // batch
#define Wn 512          // i-dimension
#define Hn 512          // j-dimension
#define Un 49           // taps
#define KWn 7
#define MP 112          // batch padded to 7 tiles of 16
#define DP_STRIDE 576   // padded diag row stride (>= 511 + 52, 64-aligned)
#define KT 52           // K padded to 13 * 4

// Kernel 1: extract diagonal of x into a zero-padded [MP][DP_STRIDE] matrix in ws.
// dp[m][c] = x[m, c, c, 0] for m<100, c<512; zero elsewhere (incl. pad rows/cols).
__global__ void build_diag_pad(const float* __restrict__ x, float* __restrict__ dp) {
    int idx = blockIdx.x * blockDim.x + threadIdx.x;
    if (idx >= MP * DP_STRIDE) return;
    int m = idx / DP_STRIDE;
    int c = idx - m * DP_STRIDE;
    float v = 0.0f;
    if (m < Bn && c < Hn)
        v = x[(size_t)m * (Wn * (size_t)Hn) + (size_t)c * Hn + c];
    dp[idx] = v;
}

// Kernel 2: per (j, i_tile) block: out[b, i, j] = 49 * sum_t dp[b][j+t] * w_eff[i][j][t] + bsum[i]
// 7 waves, each computes one 16(batch) x 16(i) tile via 13x v_wmma_f32_16x16x4_f32.
__global__ __launch_bounds__(224) void conv_diag_wmma(
    const float* __restrict__ w, const float* __restrict__ bias,
    const float* __restrict__ dp, float* __restrict__ out)
{
    __shared__ float w_tile[KT][16];  // masked weights, K-major (row t, col n)
    __shared__ float bsum[16];

    const int i_base = blockIdx.x << 4;  // 16 consecutive i's
    const int j = blockIdx.y;
    const int tid = threadIdx.x;
    const int lane = tid & 31;
    const int wv = tid >> 5;             // wave id = batch tile 0..6

    // ---- stage masked weight slice into LDS (read from HBM exactly once) ----
    if (tid < 16) bsum[tid] = 0.0f;
    for (int idx = tid; idx < KT * 16; idx += 224) {
        int t = idx >> 4;
        int n = idx & 15;
        int i = i_base + n;
        float v = 0.0f;
        if (t < Un && (i + t / KWn) < Wn && (j + t) < Hn)
            v = w[(size_t)(i * Hn + j) * Un + t];
        w_tile[t][n] = v;
    }
    __syncthreads();

    // ---- bias reduction: 224 threads = 16 i x 14 chunks of 4 taps ----
    {
        int n = tid & 15;
        int chunk = tid >> 4;            // 0..13
        int i = i_base + n;
        float s = 0.0f;
        int t0 = chunk * 4;
        #pragma unroll
        for (int dt = 0; dt < 4; ++dt) {
            int t = t0 + dt;
            if (t < Un) s += bias[(size_t)(i * Hn + j) * Un + t];
        }
        atomicAdd(&bsum[n], s);          // ds_add_f32
    }
    __syncthreads();

    // ---- WMMA accumulation over K = 52 (13 steps of 4) ----
    const int half = lane >> 4;          // 0: K even pair, 1: K odd pair (per ISA layout)
    const int sub  = lane & 15;
    v8f c = {};
    // A row for this lane: batch row m = wv*16 + sub, columns j+t of dp
    const float* dprow = dp + (size_t)(wv * 16 + sub) * DP_STRIDE + j;
    #pragma unroll
    for (int k = 0; k < KT / 4; ++k) {
        int trow = k * 4 + half * 2;
        v2f a, b;
        // A 16x4 fp32 layout: lanes 0-15: K = {4k, 4k+1}; lanes 16-31: K = {4k+2, 4k+3}
        a.x = dprow[trow];
        a.y = dprow[trow + 1];
        // B 4x16 fp32 layout: lanes 0-15 hold rows {4k, 4k+1}; lanes 16-31 rows {4k+2, 4k+3}
        b.x = w_tile[trow][sub];
        b.y = w_tile[trow + 1][sub];
        c = __builtin_amdgcn_wmma_f32_16x16x4_f32(
                /*neg_a=*/false, a, /*neg_b=*/false, b,
                /*c_mod=*/(short)0, c, /*reuse_a=*/false, /*reuse_b=*/false);
    }

    // ---- store: D VGPR r -> M = r + 8*half (per ISA C/D layout), N = sub ----
    const float bs = bsum[sub];
    const size_t icol = (size_t)(i_base + sub) * Hn + j;
    #pragma unroll
    for (int r = 0; r < 8; ++r) {
        int brow = wv * 16 + half * 8 + r;
        if (brow < Bn)
            out[(size_t)brow * (Wn * (size_t)Hn) + icol] = 49.0f * c[r] + bs;
    }
}

extern "C" void kernel_launch(void* const* d_in, const int* in_sizes, int n_in,
                              void* d_out, int out_size, void* d_ws, size_t ws_size,
                              hipStream_t stream) {
    const float* x    = (const float*)d_in[0];
    const float* w    = (const float*)d_in[1];
    const float* bias = (const float*)d_in[2];
    float* out = (float*)d_out;
    float* dp  = (float*)d_ws;          // needs MP*DP_STRIDE*4 = ~0.25 MB

    const int n1 = MP * DP_STRIDE;
    build_diag_pad<<<(n1 + 255) / 256, 256, 0, stream>>>(x, dp);

    dim3 grid(Wn / 16, Hn);             // (i_tile, j)
    conv_diag_wmma<<<grid, 224, 0, stream>>>(w, bias, dp, out);
}